// s_prod_KernelChannels5_64226940944733
// MI455X (gfx1250) — compile-verified
//
#include <hip/hip_runtime.h>
#include <stdint.h>

// ---------------------------------------------------------------------------
// s_prod 5x5 propagation stencil for MI455X (gfx1250).
// Bandwidth-bound: 8 MiB in, 100 MiB out -> ~4.9 us at 23.3 TB/s.
// CDNA5 paths used: async global->LDS halo staging (ASYNCcnt +
// s_wait_asynccnt), b128 LDS reads, non-temporal b128 global stores.
// ---------------------------------------------------------------------------

typedef float v4f __attribute__((ext_vector_type(4)));
typedef int   v2i __attribute__((vector_size(8)));   // matches builtin param

typedef __attribute__((address_space(1))) v2i as1_v2i;
typedef __attribute__((address_space(3))) v2i as3_v2i;

#define IMG     512
#define HW      (512 * 512)
#define TILE_W  64
#define TILE_H  16
#define LDSW    72              // 68 used + pad
#define LDSH    20              // 16 + 2*2 halo

__device__ __forceinline__ void async_copy_b64(const float* gsrc, float* ldst) {
#if __has_builtin(__builtin_amdgcn_global_load_async_to_lds_b64)
  __builtin_amdgcn_global_load_async_to_lds_b64(
      (as1_v2i*)gsrc, (as3_v2i*)ldst, 0, 0);
#else
  asm volatile("global_load_async_to_lds_b64 %0, %1, off"
               :: "v"((__attribute__((address_space(3))) void*)ldst),
                  "v"((__attribute__((address_space(1))) void*)gsrc)
               : "memory");
#endif
}

__device__ __forceinline__ void wait_async0() {
#if __has_builtin(__builtin_amdgcn_s_wait_asynccnt)
  __builtin_amdgcn_s_wait_asynccnt(0);
#else
  asm volatile("s_wait_asynccnt 0" ::: "memory");
#endif
}

__global__ __launch_bounds__(256)
void s_prod_stencil_kernel(const float* __restrict__ s,
                           const float* __restrict__ cs,
                           float* __restrict__ out) {
  __shared__ __align__(16) float lds[2 * LDSW * LDSH];
  float* lds_s = lds;
  float* lds_c = lds + LDSW * LDSH;

  const int tid = threadIdx.x;
  const int b   = blockIdx.z;
  const int gx0 = blockIdx.x * TILE_W;
  const int gy0 = blockIdx.y * TILE_H;

  // ---- zero-fill LDS (provides the zero padding at image borders) ----
  {
    v4f z = {0.f, 0.f, 0.f, 0.f};
    for (int i = tid; i < (2 * LDSW * LDSH) / 4; i += 256)
      ((v4f*)lds)[i] = z;
  }
  __syncthreads();

  const float* sbase = s  + (size_t)b * HW;
  const float* cbase = cs + (size_t)b * HW;

  // ---- async halo staging: 34 even-aligned b64 pairs x 20 rows, per array ----
  for (int idx = tid; idx < 34 * 20; idx += 256) {
    const int row  = idx / 34;
    const int px   = idx - row * 34;
    const int grow = gy0 - 2 + row;
    const int gcol = gx0 - 2 + 2 * px;   // even -> pairs never straddle 0/512
    if ((unsigned)grow < 512u && (unsigned)gcol < 512u) {
      const int loff = row * LDSW + 2 * px;
      async_copy_b64(sbase + (size_t)grow * IMG + gcol, lds_s + loff);
      async_copy_b64(cbase + (size_t)grow * IMG + gcol, lds_c + loff);
    }
  }
  wait_async0();
  __syncthreads();

  // ---- per-thread register window: 4 consecutive pixels ----
  const int tx = tid & 15;        // 16 threads * 4 px = 64 px wide
  const int ty = tid >> 4;        // 16 rows
  const int r0 = ty;              // LDS row of window top (center row - 2)
  const int c0 = tx * 4;          // LDS col of window left (16B aligned)

  float ws[5][8];                 // s: rows h-2..h+2, cols w0-2..w0+5
  float wc[3][8];                 // cs: rows h-1..h+1 only (confidences)
#pragma unroll
  for (int r = 0; r < 5; ++r) {
    v4f a = *(const v4f*)&lds_s[(r0 + r) * LDSW + c0];
    v4f d = *(const v4f*)&lds_s[(r0 + r) * LDSW + c0 + 4];
    ws[r][0] = a.x; ws[r][1] = a.y; ws[r][2] = a.z; ws[r][3] = a.w;
    ws[r][4] = d.x; ws[r][5] = d.y; ws[r][6] = d.z; ws[r][7] = d.w;
  }
#pragma unroll
  for (int r = 0; r < 3; ++r) {
    v4f a = *(const v4f*)&lds_c[(r0 + 1 + r) * LDSW + c0];
    v4f d = *(const v4f*)&lds_c[(r0 + 1 + r) * LDSW + c0 + 4];
    wc[r][0] = a.x; wc[r][1] = a.y; wc[r][2] = a.z; wc[r][3] = a.w;
    wc[r][4] = d.x; wc[r][5] = d.y; wc[r][6] = d.z; wc[r][7] = d.w;
  }

  // ---- inner-ring products: sXY = s*sr(X,Y), cXY = cs*cr(X,Y) ----
  float s11[4], s12[4], s13[4], s21[4], s23[4], s31[4], s32[4], s33[4];
  float c11[4], c12[4], c13[4], c21[4], c23[4], c31[4], c32[4], c33[4];
#pragma unroll
  for (int p = 0; p < 4; ++p) {
    const float sc = ws[2][p + 2];
    const float cc = wc[1][p + 2];
    s11[p] = sc * ws[1][p + 1];  c11[p] = cc * wc[0][p + 1];
    s12[p] = sc * ws[1][p + 2];  c12[p] = cc * wc[0][p + 2];
    s13[p] = sc * ws[1][p + 3];  c13[p] = cc * wc[0][p + 3];
    s21[p] = sc * ws[2][p + 1];  c21[p] = cc * wc[1][p + 1];
    s23[p] = sc * ws[2][p + 3];  c23[p] = cc * wc[1][p + 3];
    s31[p] = sc * ws[3][p + 1];  c31[p] = cc * wc[2][p + 1];
    s32[p] = sc * ws[3][p + 2];  c32[p] = cc * wc[2][p + 2];
    s33[p] = sc * ws[3][p + 3];  c33[p] = cc * wc[2][p + 3];
  }

  const int h  = gy0 + ty;
  const int w0 = gx0 + tx * 4;
  float* ob = out + (size_t)b * 25 * HW + (size_t)h * IMG + w0;

  // argmax(first-occurrence) selection, branchless
#define PK2(sa, sb_, ca, cb_) ((cb_[p] > ca[p]) ? sb_[p] : sa[p])
#define PK3(sa, sb_, sc_, ca, cb_, cc_)                           \
  ((ca[p] >= cb_[p]) ? ((ca[p] >= cc_[p]) ? sa[p] : sc_[p])       \
                     : ((cb_[p] >= cc_[p]) ? sb_[p] : sc_[p]))
#define STORE_K(k, EXPR) do { v4f v_;                              \
    { enum { p = 0 }; v_.x = (EXPR); }                             \
    { enum { p = 1 }; v_.y = (EXPR); }                             \
    { enum { p = 2 }; v_.z = (EXPR); }                             \
    { enum { p = 3 }; v_.w = (EXPR); }                             \
    __builtin_nontemporal_store(v_, (v4f*)(ob + (size_t)(k) * HW)); \
  } while (0)

  STORE_K(0,  s11[p] * ws[0][p + 0]);
  STORE_K(1,  PK2(s11, s12, c11, c12) * ws[0][p + 1]);
  STORE_K(2,  PK3(s11, s12, s13, c11, c12, c13) * ws[0][p + 2]);
  STORE_K(3,  PK2(s12, s13, c12, c13) * ws[0][p + 3]);
  STORE_K(4,  s13[p] * ws[0][p + 4]);
  STORE_K(5,  PK2(s11, s21, c11, c21) * ws[1][p + 0]);
  STORE_K(6,  s11[p]);
  STORE_K(7,  s12[p]);
  STORE_K(8,  s13[p]);
  STORE_K(9,  PK2(s13, s23, c13, c23) * ws[1][p + 4]);
  STORE_K(10, PK3(s11, s21, s31, c11, c21, c31) * ws[2][p + 0]);
  STORE_K(11, s21[p]);
  STORE_K(12, 1.0f);
  STORE_K(13, s23[p]);
  STORE_K(14, PK3(s13, s23, s33, c13, c23, c33) * ws[2][p + 4]);
  STORE_K(15, PK2(s21, s31, c21, c31) * ws[3][p + 0]);
  STORE_K(16, s31[p]);
  STORE_K(17, s32[p]);
  STORE_K(18, s33[p]);
  STORE_K(19, PK2(s23, s33, c23, c33) * ws[3][p + 4]);
  STORE_K(20, s31[p] * ws[4][p + 0]);
  STORE_K(21, PK2(s31, s32, c31, c32) * ws[4][p + 1]);
  STORE_K(22, PK3(s31, s32, s33, c31, c32, c33) * ws[4][p + 2]);
  STORE_K(23, PK2(s32, s33, c32, c33) * ws[4][p + 3]);
  STORE_K(24, s33[p] * ws[4][p + 4]);

#undef STORE_K
#undef PK3
#undef PK2
}

extern "C" void kernel_launch(void* const* d_in, const int* in_sizes, int n_in,
                              void* d_out, int out_size, void* d_ws, size_t ws_size,
                              hipStream_t stream) {
  (void)in_sizes; (void)n_in; (void)out_size; (void)d_ws; (void)ws_size;
  const float* s  = (const float*)d_in[0];
  const float* cs = (const float*)d_in[1];
  float* out      = (float*)d_out;

  dim3 grid(IMG / TILE_W, IMG / TILE_H, 4);   // 8 x 32 x 4 = 1024 blocks
  dim3 block(256, 1, 1);                      // 8 wave32 per block
  hipLaunchKernelGGL(s_prod_stencil_kernel, grid, block, 0, stream, s, cs, out);
}